// aspect_augumentation_movie_18511309046226
// MI455X (gfx1250) — compile-verified
//
#include <hip/hip_runtime.h>
#include <hip/hip_bf16.h>

typedef float v2f __attribute__((ext_vector_type(2)));
typedef float v8f __attribute__((ext_vector_type(8)));

// ---------------------------------------------------------------------------
// Zero the sums/counts workspace (harness poisons d_ws with 0xAA every run).
// ---------------------------------------------------------------------------
__global__ void zero_ws_kernel(float* __restrict__ ws, int n) {
  int i = blockIdx.x * blockDim.x + threadIdx.x;
  if (i < n) ws[i] = 0.0f;
}

// ---------------------------------------------------------------------------
// scores = softmax(leaky_relu(users @ relation_k, 0.2), axis=1)
// One wave handles 16 users via chained V_WMMA_F32_16X16X4_F32 (K=64 in 16
// steps). relation_k is zero-padded from 3 to 16 columns inside the B-frag:
// the pad is done branch-free (clamped in-bounds load * lane mask) so the
// compiler emits plain loads + v_cndmask instead of EXEC-masked branches.
// A-frag (16x4 f32): lane m (0..15) holds row m K={kk,kk+1}; lanes 16..31
// hold K={kk+2,kk+3}. B-frag (4x16): row K striped across lanes 0..15,
// second lane-half holds rows K+2/K+3.
// ---------------------------------------------------------------------------
__global__ __launch_bounds__(256) void scores_kernel(
    const int*   __restrict__ user_id,
    const float* __restrict__ user_factors,   // [N_USERS, 64]
    const float* __restrict__ relation_k,     // [64, 3] row-major
    float*       __restrict__ scores_out,     // [B, 3]
    int B)
{
  __shared__ float sdata[8][32][8];           // 8 waves * 32 lanes * 8 acc regs
  const int wave = threadIdx.x >> 5;
  const int lane = threadIdx.x & 31;
  const int gw   = blockIdx.x * 8 + wave;     // global wave id
  const int userBase = gw * 16;

  // Keep EXEC all-ones through the WMMA sequence: clamp instead of returning.
  const int m    = lane & 15;                 // matrix row this lane feeds
  int urow = userBase + m;
  if (urow >= B) urow = B - 1;
  const int uid = user_id[urow];
  const float* rowptr = user_factors + (size_t)uid * 64;
  const int koff = (lane < 16) ? 0 : 2;
  const int n      = m;                       // B-matrix column for this lane
  const int ncl    = (n < 3) ? n : 2;         // clamped, always in-bounds
  const float nmsk = (n < 3) ? 1.0f : 0.0f;   // zero pad via VALU select

  v8f acc = {0.f, 0.f, 0.f, 0.f, 0.f, 0.f, 0.f, 0.f};
#pragma unroll
  for (int kk = 0; kk < 64; kk += 4) {
    v2f a;
    a.x = rowptr[kk + koff + 0];
    a.y = rowptr[kk + koff + 1];
    v2f b;
    b.x = relation_k[(kk + koff + 0) * 3 + ncl] * nmsk;
    b.y = relation_k[(kk + koff + 1) * 3 + ncl] * nmsk;
    acc = __builtin_amdgcn_wmma_f32_16x16x4_f32(
        /*neg_a=*/false, a, /*neg_b=*/false, b,
        /*c_mod=*/(short)0, acc, /*reuse_a=*/false, /*reuse_b=*/false);
  }

  // Spill D to LDS so each of lanes 0..15 can collect its user's 3 logits.
#pragma unroll
  for (int r = 0; r < 8; ++r) sdata[wave][lane][r] = acc[r];
  __syncthreads();

  if (lane < 16 && (userBase + lane) < B) {
    const int u = lane;  // user index within tile
    float l[3];
#pragma unroll
    for (int j = 0; j < 3; ++j) {
      // D[M=u][N=j]: u<8 -> VGPR u, lane j ; u>=8 -> VGPR u-8, lane j+16
      l[j] = (u < 8) ? sdata[wave][j][u] : sdata[wave][j + 16][u - 8];
      // leaky_relu slope 0.2
      l[j] = (l[j] > 0.0f) ? l[j] : 0.2f * l[j];
    }
    float mx = fmaxf(l[0], fmaxf(l[1], l[2]));
    float e0 = __expf(l[0] - mx);
    float e1 = __expf(l[1] - mx);
    float e2 = __expf(l[2] - mx);
    float inv = 1.0f / (e0 + e1 + e2);
    float* o = scores_out + (size_t)(userBase + u) * 3;
    o[0] = e0 * inv;
    o[1] = e1 * inv;
    o[2] = e2 * inv;
  }
}

// ---------------------------------------------------------------------------
// Two edges per wave: each 16-lane half loads one 64-float entity row and one
// user row with a single float4 (global_load_b128) per matrix — a fully
// coalesced 256B burst. 4-step xor-shuffle reduction stays within each half
// (offsets <= 8). Lane 0 of each half atomically accumulates sum and count.
// Memory-bound by design; compute is free at 23.3 TB/s.
// ---------------------------------------------------------------------------
__global__ __launch_bounds__(256) void edge_dot_kernel(
    const int*   __restrict__ idx,
    const int*   __restrict__ seg,
    int nE,
    const int*   __restrict__ user_id,
    const float* __restrict__ user_factors,
    const float* __restrict__ entity_factors,
    float*       __restrict__ sums,
    float*       __restrict__ cnts)
{
  const int lane = threadIdx.x & 31;
  const int half = lane >> 4;                  // which edge in this wave
  const int l    = lane & 15;
  const int gw = (blockIdx.x * blockDim.x + threadIdx.x) >> 5;
  const int nW = (gridDim.x * blockDim.x) >> 5;
  for (int eb = gw * 2; eb < nE; eb += nW * 2) {
    const int e = eb + half;
    float p = 0.0f;
    int sb = 0;
    if (e < nE) {
      const int ei  = idx[e];
      sb            = seg[e];
      const int uid = user_id[sb];
      const float4* ep = (const float4*)(entity_factors + (size_t)ei * 64);
      const float4* up = (const float4*)(user_factors  + (size_t)uid * 64);
      float4 x = ep[l];
      float4 y = up[l];
      p = x.x * y.x + x.y * y.y + x.z * y.z + x.w * y.w;
    }
#pragma unroll
    for (int off = 8; off; off >>= 1) p += __shfl_xor(p, off, 32);
    if (l == 0 && e < nE) {
      atomicAdd(&sums[sb], p);
      atomicAdd(&cnts[sb], 1.0f);
    }
  }
}

// ---------------------------------------------------------------------------
// ca/cd/cg = sums / max(cnt,1);  prediction = (ca*s0+cd*s1+cg*s2)/(s0+s1+s2)
// ---------------------------------------------------------------------------
__global__ void finalize_kernel(
    const float* __restrict__ ws,       // [6*B]: sumsA,cntA,sumsD,cntD,sumsG,cntG
    const float* __restrict__ scores,   // [B,3]
    float* __restrict__ pred,
    float* __restrict__ ca_o,
    float* __restrict__ cd_o,
    float* __restrict__ cg_o,
    int B)
{
  int b = blockIdx.x * blockDim.x + threadIdx.x;
  if (b >= B) return;
  float ca = ws[0 * B + b] / fmaxf(ws[1 * B + b], 1.0f);
  float cd = ws[2 * B + b] / fmaxf(ws[3 * B + b], 1.0f);
  float cg = ws[4 * B + b] / fmaxf(ws[5 * B + b], 1.0f);
  float s0 = scores[(size_t)b * 3 + 0];
  float s1 = scores[(size_t)b * 3 + 1];
  float s2 = scores[(size_t)b * 3 + 2];
  pred[b] = (ca * s0 + cd * s1 + cg * s2) / (s0 + s1 + s2);
  ca_o[b] = ca;
  cd_o[b] = cd;
  cg_o[b] = cg;
}

extern "C" void kernel_launch(void* const* d_in, const int* in_sizes, int n_in,
                              void* d_out, int out_size, void* d_ws, size_t ws_size,
                              hipStream_t stream) {
  const int*   user_id        = (const int*)d_in[0];
  const int*   actors_idx     = (const int*)d_in[1];
  const int*   actors_seg     = (const int*)d_in[2];
  const int*   directors_idx  = (const int*)d_in[3];
  const int*   directors_seg  = (const int*)d_in[4];
  const int*   genres_idx     = (const int*)d_in[5];
  const int*   genres_seg     = (const int*)d_in[6];
  const float* user_factors   = (const float*)d_in[7];
  const float* entity_factors = (const float*)d_in[8];
  const float* relation_k     = (const float*)d_in[9];

  const int B  = in_sizes[0];
  const int nA = in_sizes[1];
  const int nD = in_sizes[3];
  const int nG = in_sizes[5];

  float* ws  = (float*)d_ws;          // 6*B floats: sums/cnts for a/d/g
  float* out = (float*)d_out;
  float* pred   = out;                // [B]
  float* scores = out + (size_t)B;    // [B,3]
  float* ca_o   = out + (size_t)4 * B;
  float* cd_o   = out + (size_t)5 * B;
  float* cg_o   = out + (size_t)6 * B;

  // 1) zero sums/counts
  {
    int n = 6 * B;
    zero_ws_kernel<<<(n + 255) / 256, 256, 0, stream>>>(ws, n);
  }

  // 2) scores via WMMA (16 users per wave, 8 waves per block)
  {
    int blocks = (B + 127) / 128;     // 128 users per 256-thread block
    scores_kernel<<<blocks, 256, 0, stream>>>(user_id, user_factors,
                                              relation_k, scores, B);
  }

  // 3) segment dot-product passes (2 edges per wave, 16 edges per block)
  {
    int blocksA = (nA + 15) / 16;
    edge_dot_kernel<<<blocksA, 256, 0, stream>>>(actors_idx, actors_seg, nA,
        user_id, user_factors, entity_factors, ws + 0 * (size_t)B, ws + 1 * (size_t)B);
    int blocksD = (nD + 15) / 16;
    edge_dot_kernel<<<blocksD, 256, 0, stream>>>(directors_idx, directors_seg, nD,
        user_id, user_factors, entity_factors, ws + 2 * (size_t)B, ws + 3 * (size_t)B);
    int blocksG = (nG + 15) / 16;
    edge_dot_kernel<<<blocksG, 256, 0, stream>>>(genres_idx, genres_seg, nG,
        user_id, user_factors, entity_factors, ws + 4 * (size_t)B, ws + 5 * (size_t)B);
  }

  // 4) finalize means + prediction
  finalize_kernel<<<(B + 255) / 256, 256, 0, stream>>>(ws, scores, pred,
                                                       ca_o, cd_o, cg_o, B);
}